// ThetaRestraint_81612968558777
// MI455X (gfx1250) — compile-verified
//
#include <hip/hip_runtime.h>
#include <math.h>

#define LL    500
#define NBINS 24
#define TPB   256

// ---- CDNA5 async global->LDS copy, 16 bytes per enabled lane (ASYNCcnt) ----
__device__ __forceinline__ void async_ld_b128(unsigned lds_off, const void* g) {
    asm volatile("global_load_async_to_lds_b128 %0, %1, off"
                 :: "v"(lds_off), "v"((unsigned long long)(size_t)g)
                 : "memory");
}
__device__ __forceinline__ void wait_async0() {
    asm volatile("s_wait_asynccnt 0" ::: "memory");
}

__global__ void __launch_bounds__(TPB)
theta_restraint_kernel(const float* __restrict__ N,
                       const float* __restrict__ CA,
                       const float* __restrict__ CB,
                       const float* __restrict__ coeff,
                       const int*   __restrict__ x_idx,
                       const int*   __restrict__ y_idx,
                       int E,
                       float* __restrict__ partials)
{
    __shared__ __align__(16) float sN [LL * 3];
    __shared__ __align__(16) float sCA[LL * 3];
    __shared__ __align__(16) float sCB[LL * 3];
    __shared__ float red[TPB];

    const int tid = threadIdx.x;
    const int b   = blockIdx.y;          // batch handled by this block

    // ---- stage this batch's geometry into LDS via async b128 copies ----
    const size_t batchOff = (size_t)b * (LL * 3);
    const float* gN  = N  + batchOff;    // 6000 B, 16B-aligned (6000 % 16 == 0)
    const float* gCA = CA + batchOff;
    const float* gCB = CB + batchOff;
    // low 32 bits of a generic pointer to LDS == wave-relative LDS byte offset
    const unsigned lN  = (unsigned)(size_t)&sN [0];
    const unsigned lCA = (unsigned)(size_t)&sCA[0];
    const unsigned lCB = (unsigned)(size_t)&sCB[0];
    const int n128 = (LL * 3) / 4;       // 375 16-byte packets per array
    for (int i = tid; i < n128; i += TPB) {
        async_ld_b128(lN  + i * 16, gN  + i * 4);
        async_ld_b128(lCA + i * 16, gCA + i * 4);
        async_ld_b128(lCB + i * 16, gCB + i * 4);
    }
    wait_async0();        // this wave's async transfers landed in LDS
    __syncthreads();      // whole block's staging visible

    // ---- per-edge dihedral + spline evaluation ----
    float acc = 0.0f;
    const int e = blockIdx.x * TPB + tid;
    if (e < E) {
        const int x = x_idx[e];
        const int y = y_idx[e];

        // coeff[x][y] row: 24 bins * float4 = 384 B; prefetch all 3 cachelines
        const size_t rowIdx = ((size_t)x * LL + y) * NBINS;    // in float4 units
        const float* crow = coeff + rowIdx * 4;
        __builtin_prefetch(crow,      0, 0);   // -> global_prefetch_b8
        __builtin_prefetch(crow + 32, 0, 0);   // +128 B
        __builtin_prefetch(crow + 64, 0, 0);   // +256 B

        const int x3 = 3 * x, y3 = 3 * y;
        const float xNx  = sN [x3],     xNy  = sN [x3 + 1], xNz  = sN [x3 + 2];
        const float xCAx = sCA[x3],     xCAy = sCA[x3 + 1], xCAz = sCA[x3 + 2];
        const float xCBx = sCB[x3],     xCBy = sCB[x3 + 1], xCBz = sCB[x3 + 2];
        const float yCBx = sCB[y3],     yCBy = sCB[y3 + 1], yCBz = sCB[y3 + 2];

        const float b1x = xCAx - xNx,  b1y = xCAy - xNy,  b1z = xCAz - xNz;
        const float b2x = xCBx - xCAx, b2y = xCBy - xCAy, b2z = xCBz - xCAz;
        const float b3x = yCBx - xCBx, b3y = yCBy - xCBy, b3z = yCBz - xCBz;

        // n1 = b1 x b2 ; n2 = b2 x b3
        const float n1x = b1y * b2z - b1z * b2y;
        const float n1y = b1z * b2x - b1x * b2z;
        const float n1z = b1x * b2y - b1y * b2x;
        const float n2x = b2y * b3z - b2z * b3y;
        const float n2y = b2z * b3x - b2x * b3z;
        const float n2z = b2x * b3y - b2y * b3x;

        const float invb2 = rsqrtf(b2x * b2x + b2y * b2y + b2z * b2z);
        // m = (n1 x b2) * invb2
        const float mx = (n1y * b2z - n1z * b2y) * invb2;
        const float my = (n1z * b2x - n1x * b2z) * invb2;
        const float mz = (n1x * b2y - n1y * b2x) * invb2;

        const float sdot = mx * n2x + my * n2y + mz * n2z;
        const float cdot = n1x * n2x + n1y * n2y + n1z * n2z;
        const float theta = atan2f(sdot, cdot);

        const float PI_F = 3.14159265358979323846f;
        const float STEP = PI_F / 12.0f;                 // 15 degrees
        const float C0   = -PI_F + 0.5f * STEP;          // first cutoff
        const float tw = (theta < C0) ? theta + 2.0f * PI_F : theta;

        int seg = (int)((tw - C0) * (12.0f / PI_F));     // uniform-bin searchsorted
        seg = seg < 0 ? 0 : (seg > NBINS - 1 ? NBINS - 1 : seg);
        const float t = tw - (C0 + (float)seg * STEP);

        const float4 cs = ((const float4*)coeff)[rowIdx + seg];  // 16B L2 gather
        acc = ((cs.w * t + cs.z) * t + cs.y) * t + cs.x;
    }

    // ---- deterministic block reduction ----
    red[tid] = acc;
    __syncthreads();
    #pragma unroll
    for (int s = TPB / 2; s > 0; s >>= 1) {
        if (tid < s) red[tid] += red[tid + s];
        __syncthreads();
    }
    if (tid == 0)
        partials[(size_t)blockIdx.y * gridDim.x + blockIdx.x] = red[0];
}

// fixed-order final sum -> deterministic across graph replays
__global__ void theta_restraint_finish(const float* __restrict__ partials,
                                       int n, float* __restrict__ out)
{
    if (blockIdx.x == 0 && threadIdx.x == 0) {
        float s = 0.0f;
        for (int i = 0; i < n; ++i) s += partials[i];
        out[0] = s;
    }
}

extern "C" void kernel_launch(void* const* d_in, const int* in_sizes, int n_in,
                              void* d_out, int out_size, void* d_ws, size_t ws_size,
                              hipStream_t stream)
{
    const float* N     = (const float*)d_in[0];
    const float* CA    = (const float*)d_in[1];
    const float* CB    = (const float*)d_in[2];
    const float* coeff = (const float*)d_in[3];
    const int*   x_idx = (const int*)d_in[4];
    const int*   y_idx = (const int*)d_in[5];

    const int E = in_sizes[4];                 // number of edges
    const int B = in_sizes[0] / (LL * 3);      // batches (== 4)

    const int nbx = (E + TPB - 1) / TPB;       // 391 blocks along edges
    dim3 grid(nbx, B);

    float* partials = (float*)d_ws;            // nbx*B floats of scratch

    theta_restraint_kernel<<<grid, TPB, 0, stream>>>(N, CA, CB, coeff,
                                                     x_idx, y_idx, E, partials);
    theta_restraint_finish<<<1, 32, 0, stream>>>(partials, nbx * B, (float*)d_out);
}